// InterSuperpixelPCR_15556371546820
// MI455X (gfx1250) — compile-verified
//
#include <hip/hip_runtime.h>
#include <hip/hip_bf16.h>
#include <math.h>

typedef __attribute__((ext_vector_type(2))) float v2f;
typedef __attribute__((ext_vector_type(8))) float v8f;

#define B_   2
#define P_   16
#define HW_  (1024*1024)
#define K_   1000
#define KPAD 1008   // M padded to multiple of 16 for WMMA tiles

// ---- workspace layout (in floats) ----
#define OFF_SUMS   0                       // B*K*P = 32000
#define OFF_CNT    32000                   // K     = 1000
#define OFF_FEATS  33000                   // B*K*P = 32000
#define OFF_ATTW   65536                   // KPAD*K = 1,008,000 (padded softmax)
#define OFF_SM     1073536                 // B*K*P = 32000 (smoothed features)

// ------------------------------------------------------------------
// 0) zero sums + counts
__global__ void zero_ws(float* ws, int n) {
    int i = blockIdx.x * blockDim.x + threadIdx.x;
    int stride = gridDim.x * blockDim.x;
    for (; i < n; i += stride) ws[i] = 0.0f;
}

// ------------------------------------------------------------------
// 1) segment counts, staged in LDS (4KB), flushed with global atomics
__global__ void seg_counts(const int* __restrict__ seg, float* __restrict__ cnt) {
    __shared__ float lcnt[K_];
    int tid = threadIdx.x;
    for (int j = tid; j < K_; j += blockDim.x) lcnt[j] = 0.0f;
    __syncthreads();
    int i = blockIdx.x * blockDim.x + tid;
    int stride = gridDim.x * blockDim.x;
    for (; i < HW_; i += stride) atomicAdd(&lcnt[seg[i]], 1.0f);
    __syncthreads();
    for (int j = tid; j < K_; j += blockDim.x) {
        float v = lcnt[j];
        if (v != 0.0f) atomicAdd(&cnt[j], v);
    }
}

// ------------------------------------------------------------------
// 2) segment sums per batch, staged in LDS (64KB = K*P floats).
//    LDS layout [p*K + s] keeps consecutive lanes (different s) in
//    different banks for each fixed p.
__global__ void seg_sums(const float* __restrict__ fused,
                         const int* __restrict__ seg,
                         float* __restrict__ sums) {
    __shared__ float lsum[K_ * P_];
    int tid = threadIdx.x;
    int b = blockIdx.y;
    for (int j = tid; j < K_ * P_; j += blockDim.x) lsum[j] = 0.0f;
    __syncthreads();
    int i = blockIdx.x * blockDim.x + tid;
    int stride = gridDim.x * blockDim.x;
    for (; i < HW_; i += stride) {
        int s = seg[i];
        #pragma unroll
        for (int p = 0; p < P_; ++p) {
            float v = fused[((size_t)(b * P_ + p) << 20) + i];
            atomicAdd(&lsum[p * K_ + s], v);
        }
    }
    __syncthreads();
    for (int j = tid; j < K_ * P_; j += blockDim.x) {
        int p = j / K_, s = j - p * K_;
        float v = lsum[j];
        if (v != 0.0f)
            atomicAdd(&sums[(size_t)b * (K_ * P_) + s * P_ + p], v);
    }
}

// ------------------------------------------------------------------
// 3) means: feats = cnt>0 ? sums/max(cnt,1) : 0   (same [b][k][p] layout)
__global__ void seg_means(const float* __restrict__ sums,
                          const float* __restrict__ cnt,
                          float* __restrict__ feats) {
    int i = blockIdx.x * blockDim.x + threadIdx.x;
    if (i >= B_ * K_ * P_) return;
    int k = (i % (K_ * P_)) / P_;
    float c = cnt[k];
    feats[i] = (c > 0.0f) ? sums[i] / fmaxf(c, 1.0f) : 0.0f;
}

// ------------------------------------------------------------------
// 4) row softmax of attention into a KPAD x K buffer (rows >= K zeroed)
__global__ void softmax_rows(const float* __restrict__ attn,
                             float* __restrict__ attw) {
    __shared__ float red[256];
    int r = blockIdx.x;
    int tid = threadIdx.x;
    float* outrow = attw + (size_t)r * K_;
    if (r >= K_) {
        for (int c = tid; c < K_; c += blockDim.x) outrow[c] = 0.0f;
        return;
    }
    const float* inrow = attn + (size_t)r * K_;
    // max
    float m = -INFINITY;
    for (int c = tid; c < K_; c += blockDim.x) m = fmaxf(m, inrow[c]);
    red[tid] = m; __syncthreads();
    for (int s = 128; s > 0; s >>= 1) {
        if (tid < s) red[tid] = fmaxf(red[tid], red[tid + s]);
        __syncthreads();
    }
    float mx = red[0]; __syncthreads();
    // sum of exp
    float sum = 0.0f;
    for (int c = tid; c < K_; c += blockDim.x) sum += __expf(inrow[c] - mx);
    red[tid] = sum; __syncthreads();
    for (int s = 128; s > 0; s >>= 1) {
        if (tid < s) red[tid] += red[tid + s];
        __syncthreads();
    }
    float inv = 1.0f / red[0];
    for (int c = tid; c < K_; c += blockDim.x)
        outrow[c] = __expf(inrow[c] - mx) * inv;
}

// ------------------------------------------------------------------
// 5) smoothed[b,m,p] = sum_l attw[m,l] * feats[b,l,p]
//    One wave per 16x16 output tile, V_WMMA_F32_16X16X4_F32, K-loop of 250.
//    ISA f32 layouts: A lane=M(lane&15), K pair = 2*(lane>>4)+{0,1};
//                     B lane=N(lane&15), K pair = 2*(lane>>4)+{0,1};
//                     D vgpr r -> M = r + 8*(lane>>4), N = lane&15.
__global__ __launch_bounds__(32)
void gemm_wmma(const float* __restrict__ attw,
               const float* __restrict__ feats,
               float* __restrict__ smoothed) {
    int lane = threadIdx.x;
    int m0 = blockIdx.x * 16;
    int b  = blockIdx.y;
    int n    = lane & 15;
    int half = lane >> 4;

    const float* Arow = attw + (size_t)(m0 + (lane & 15)) * K_;
    const float* F    = feats + (size_t)b * (K_ * P_);

    v8f acc = {};
    for (int l = 0; l < K_; l += 4) {
        int ka = l + 2 * half;
        v2f a, bb;
        a.x  = Arow[ka];
        a.y  = Arow[ka + 1];
        bb.x = F[(ka + 0) * P_ + n];
        bb.y = F[(ka + 1) * P_ + n];
        acc = __builtin_amdgcn_wmma_f32_16x16x4_f32(
            /*neg_a=*/false, a, /*neg_b=*/false, bb,
            /*c_mod=*/(short)0, acc, /*reuse_a=*/false, /*reuse_b=*/false);
    }
    #pragma unroll
    for (int r = 0; r < 8; ++r) {
        int m = m0 + r + 8 * half;
        if (m < K_)
            smoothed[((size_t)b * K_ + m) * P_ + n] = acc[r];
    }
}

// ------------------------------------------------------------------
// 6) gather + blend: out = a*smoothed[b,seg,p] + (1-a)*fused
//    One thread per pixel; reads seg once, streams all 32 (b,p) planes.
__global__ void blend(const float* __restrict__ fused,
                      const int* __restrict__ seg,
                      const float* __restrict__ smoothed,
                      const float* __restrict__ alpha,
                      float* __restrict__ out) {
    float a = fminf(fmaxf(alpha[0], 0.0f), 1.0f);
    float na = 1.0f - a;
    int i = blockIdx.x * blockDim.x + threadIdx.x;
    int stride = gridDim.x * blockDim.x;
    for (; i < HW_; i += stride) {
        int s = seg[i];
        #pragma unroll
        for (int b = 0; b < B_; ++b) {
            const float* sm = smoothed + ((size_t)b * K_ + s) * P_;
            #pragma unroll
            for (int p = 0; p < P_; ++p) {
                size_t idx = ((size_t)(b * P_ + p) << 20) + i;
                out[idx] = a * sm[p] + na * fused[idx];
            }
        }
    }
}

// ------------------------------------------------------------------
extern "C" void kernel_launch(void* const* d_in, const int* in_sizes, int n_in,
                              void* d_out, int out_size, void* d_ws, size_t ws_size,
                              hipStream_t stream) {
    const float* fused = (const float*)d_in[0];
    const int*   seg   = (const int*)d_in[1];
    const float* attn  = (const float*)d_in[2];
    const float* alpha = (const float*)d_in[3];
    float* out = (float*)d_out;
    float* ws  = (float*)d_ws;

    float* sums     = ws + OFF_SUMS;
    float* cnt      = ws + OFF_CNT;
    float* feats    = ws + OFF_FEATS;
    float* attw     = ws + OFF_ATTW;
    float* smoothed = ws + OFF_SM;

    // 0) zero the atomic targets (sums + counts = 33000 floats)
    zero_ws<<<130, 256, 0, stream>>>(ws, B_ * K_ * P_ + K_);
    // 1) counts
    seg_counts<<<1024, 256, 0, stream>>>(seg, cnt);
    // 2) sums (grid.y = batch)
    seg_sums<<<dim3(512, B_), 256, 0, stream>>>(fused, seg, sums);
    // 3) means
    seg_means<<<(B_ * K_ * P_ + 255) / 256, 256, 0, stream>>>(sums, cnt, feats);
    // 4) softmax (padded rows)
    softmax_rows<<<KPAD, 256, 0, stream>>>(attn, attw);
    // 5) WMMA GEMM: 63 M-tiles x 2 batches, one wave each
    gemm_wmma<<<dim3(KPAD / 16, B_), 32, 0, stream>>>(attw, feats, smoothed);
    // 6) gather + blend
    blend<<<4096, 256, 0, stream>>>(fused, seg, smoothed, alpha, out);
}